// ThreeHeadedDecoder_67482526154854
// MI455X (gfx1250) — compile-verified
//
#include <hip/hip_runtime.h>
#include <math.h>

// Sizes from the reference
#define V_SZ 50257
#define E_SZ 512
#define H_SZ 1024
#define C_SZ 1024
#define L_SZ 1024
#define S_SZ 2048

typedef __attribute__((ext_vector_type(2))) float v2f;
typedef __attribute__((ext_vector_type(8))) float v8f;

#define GEMV_BLOCK 256
#define GEMV_MAXK 4096

// ---------------------------------------------------------------------------
// Generic row-major GEMV via V_WMMA_F32_16X16X4_F32.
// out[r] = (acc ? out[r] : 0) + sum_k W[r*K+k]*q[k] + bias[r] + bias2[r] + *sbias
// A operand: q chunk broadcast to all 16 M-rows (lane reads q[kb + (lane<16?0:2)..+1]).
// B operand: 16 weight rows x 4 k (lane reads W[rowBase+lane%16][kb + (lane<16?0:2)..+1]).
// D row 0 (lanes 0..15, first C/D VGPR) = the 16 fp32 dot products.
// ---------------------------------------------------------------------------
__global__ void wmma_gemv(const float* __restrict__ W, const float* __restrict__ q,
                          const float* __restrict__ bias, const float* __restrict__ bias2,
                          const float* __restrict__ sbias, float* __restrict__ out,
                          int R, int K, int acc, int do_tanh) {
    __shared__ float qs[GEMV_MAXK];
    for (int k = threadIdx.x; k < K; k += GEMV_BLOCK) qs[k] = q[k];
    __syncthreads();

    const int wave = threadIdx.x >> 5;
    const int lane = threadIdx.x & 31;
    const int rowBase = (blockIdx.x * (GEMV_BLOCK / 32) + wave) * 16;
    if (rowBase >= R) return;  // wave-uniform: EXEC stays all-ones for WMMA

    const int n = lane & 15;
    const int row = rowBase + n;
    const int rclamp = (row < R) ? row : (R - 1);
    const int khalf = (lane >> 4) << 1;  // 0 for lanes 0-15, 2 for lanes 16-31
    const float* wrow = W + (size_t)rclamp * K + khalf;
    const float* qrow = qs + khalf;

    v8f c = {0.f, 0.f, 0.f, 0.f, 0.f, 0.f, 0.f, 0.f};
    for (int kb = 0; kb < K; kb += 4) {
        v2f a = *(const v2f*)(qrow + kb);   // LDS b64: q[kb+khalf], q[kb+khalf+1]
        v2f b = *(const v2f*)(wrow + kb);   // global b64: W[row][kb+khalf..+1]
        c = __builtin_amdgcn_wmma_f32_16x16x4_f32(false, a, false, b,
                                                  (short)0, c, false, false);
    }

    if (lane < 16 && row < R) {
        float r = c[0];                     // D[M=0][N=lane]
        if (acc)   r += out[row];
        if (bias)  r += bias[row];
        if (bias2) r += bias2[row];
        if (sbias) r += *sbias;
        if (do_tanh) r = tanhf(r);
        out[row] = r;
    }
}

// out[c] = sum_r M[r*cols + c] * v[r]   (column matvec, coalesced across c)
__global__ void colmatvec(const float* __restrict__ M, const float* __restrict__ v,
                          float* __restrict__ out, int rows, int cols) {
    int c = blockIdx.x * blockDim.x + threadIdx.x;
    if (c >= cols) return;
    float s = 0.f;
    for (int r = 0; r < rows; ++r) s += M[(size_t)r * cols + c] * v[r];
    out[c] = s;
}

// single-block dot product, n multiple of blockDim
__global__ void dotk(const float* __restrict__ a, const float* __restrict__ b,
                     int n, float* __restrict__ out) {
    __shared__ float red[256];
    float s = 0.f;
    for (int i = threadIdx.x; i < n; i += 256) s += a[i] * b[i];
    red[threadIdx.x] = s;
    __syncthreads();
    for (int off = 128; off > 0; off >>= 1) {
        if (threadIdx.x < off) red[threadIdx.x] += red[threadIdx.x + off];
        __syncthreads();
    }
    if (threadIdx.x == 0) out[0] = red[0];
}

// in-place softmax over exactly 2048 elements, single block of 256
__global__ void softmax2048(float* __restrict__ e) {
    __shared__ float red[256];
    float loc[8];
    float m = -INFINITY;
    for (int j = 0; j < 8; ++j) {
        loc[j] = e[threadIdx.x + 256 * j];
        m = fmaxf(m, loc[j]);
    }
    red[threadIdx.x] = m;
    __syncthreads();
    for (int off = 128; off > 0; off >>= 1) {
        if (threadIdx.x < off)
            red[threadIdx.x] = fmaxf(red[threadIdx.x], red[threadIdx.x + off]);
        __syncthreads();
    }
    m = red[0];
    __syncthreads();
    float s = 0.f;
    for (int j = 0; j < 8; ++j) { loc[j] = expf(loc[j] - m); s += loc[j]; }
    red[threadIdx.x] = s;
    __syncthreads();
    for (int off = 128; off > 0; off >>= 1) {
        if (threadIdx.x < off) red[threadIdx.x] += red[threadIdx.x + off];
        __syncthreads();
    }
    float inv = 1.0f / red[0];
    for (int j = 0; j < 8; ++j) e[threadIdx.x + 256 * j] = loc[j] * inv;
}

// LSTM gate nonlinearity (torch order i,f,g,o); writes h to two places, c to one
__global__ void lstm_finish(const float* __restrict__ g, const float* __restrict__ cprev,
                            float* __restrict__ h1, float* __restrict__ h2,
                            float* __restrict__ cout) {
    int h = blockIdx.x * blockDim.x + threadIdx.x;
    if (h >= H_SZ) return;
    float i  = 1.f / (1.f + expf(-g[h]));
    float f  = 1.f / (1.f + expf(-g[H_SZ + h]));
    float gg = tanhf(g[2 * H_SZ + h]);
    float o  = 1.f / (1.f + expf(-g[3 * H_SZ + h]));
    float c  = f * cprev[h] + i * gg;
    float hh = o * tanhf(c);
    h1[h] = hh;
    h2[h] = hh;
    cout[h] = c;
}

__global__ void embed_gather(const int* __restrict__ ids, const float* __restrict__ emb,
                             float* __restrict__ x) {
    int e = blockIdx.x * blockDim.x + threadIdx.x;
    if (e < E_SZ) x[e] = emb[(size_t)ids[0] * E_SZ + e];
}

// ---------------------------------------------------------------------------
// Host launcher
// ---------------------------------------------------------------------------
extern "C" void kernel_launch(void* const* d_in, const int* in_sizes, int n_in,
                              void* d_out, int out_size, void* d_ws, size_t ws_size,
                              hipStream_t stream) {
    const int*   ids      = (const int*)  d_in[0];
    const float* h0in     = (const float*)d_in[1];   // [2,1,H]
    const float* c0in     = (const float*)d_in[2];   // [2,1,H]
    const float* ctxL     = (const float*)d_in[3];   // [S,C]
    const float* ctxR     = (const float*)d_in[4];
    const float* ctxM     = (const float*)d_in[5];
    const float* emb      = (const float*)d_in[6];
    const float* W_ih0    = (const float*)d_in[7];   // [4H,E]
    const float* W_hh0    = (const float*)d_in[8];   // [4H,H]
    const float* b_ih0    = (const float*)d_in[9];
    const float* b_hh0    = (const float*)d_in[10];
    const float* W_ih1    = (const float*)d_in[11];  // [4H,H]
    const float* W_hh1    = (const float*)d_in[12];
    const float* b_ih1    = (const float*)d_in[13];
    const float* b_hh1    = (const float*)d_in[14];
    const float* W_left   = (const float*)d_in[15];  // [H,C]
    const float* b_left   = (const float*)d_in[16];
    const float* W_right  = (const float*)d_in[17];
    const float* b_right  = (const float*)d_in[18];
    const float* W_lemma  = (const float*)d_in[19];
    const float* b_lemma  = (const float*)d_in[20];
    const float* W_concat = (const float*)d_in[21];  // [H, H+2C+L]
    const float* b_concat = (const float*)d_in[22];
    const float* W_out    = (const float*)d_in[23];  // [V,H]
    const float* b_out    = (const float*)d_in[24];

    float* out = (float*)d_out;           // [V] logits | [2H] h_new | [2H] c_new
    float* ws  = (float*)d_ws;

    // workspace layout (floats)
    float* X      = ws;                   // 512   embedded token
    float* GATES  = ws + 512;             // 4096  LSTM pre-activations
    float* Hy0    = ws + 4608;            // 1024  h_0
    float* Y      = ws + 5632;            // 4096  [h_1 | lctx | rctx | mctx]
    float* U      = ws + 9728;            // 1024  Wᵀq per head (reused)
    float* Ebuf   = ws + 10752;           // 2048  scores / softmax weights (reused)
    float* BQ     = ws + 12800;           // 1     scalar b·q (reused)
    float* COUT   = ws + 12832;           // 1024  concat_out

    float* h_new = out + V_SZ;            // h_0 then h_1
    float* c_new = out + V_SZ + 2 * H_SZ; // c_0 then c_1

    const int G4H = (4 * H_SZ) / 128;             // 32 blocks of 128 rows
    const int GS  = S_SZ / 128;                   // 16
    const int GH  = H_SZ / 128;                   // 8
    const int GV  = (V_SZ + 127) / 128;           // 393

    // x = embedding[input_ids]
    embed_gather<<<2, 256, 0, stream>>>(ids, emb, X);

    // ---- LSTM layer 0: gates = W_ih0·x + b_ih0 + b_hh0 + W_hh0·h0[0]
    wmma_gemv<<<G4H, GEMV_BLOCK, 0, stream>>>(W_ih0, X, b_ih0, b_hh0, nullptr,
                                              GATES, 4 * H_SZ, E_SZ, 0, 0);
    wmma_gemv<<<G4H, GEMV_BLOCK, 0, stream>>>(W_hh0, h0in, nullptr, nullptr, nullptr,
                                              GATES, 4 * H_SZ, H_SZ, 1, 0);
    lstm_finish<<<4, 256, 0, stream>>>(GATES, c0in, Hy0, h_new, c_new);

    // ---- LSTM layer 1
    wmma_gemv<<<G4H, GEMV_BLOCK, 0, stream>>>(W_ih1, Hy0, b_ih1, b_hh1, nullptr,
                                              GATES, 4 * H_SZ, H_SZ, 0, 0);
    wmma_gemv<<<G4H, GEMV_BLOCK, 0, stream>>>(W_hh1, h0in + H_SZ, nullptr, nullptr, nullptr,
                                              GATES, 4 * H_SZ, H_SZ, 1, 0);
    // h_1 lands at Y[0:1024] (start of concat vector) and in d_out
    lstm_finish<<<4, 256, 0, stream>>>(GATES, c0in + H_SZ, Y, h_new + H_SZ, c_new + H_SZ);

    // ---- three attention heads: e = ctx·(Wᵀh₁) + (b·h₁); a = softmax(e); out = aᵀ·ctx
    const float* headW[3]   = {W_left, W_right, W_lemma};
    const float* headB[3]   = {b_left, b_right, b_lemma};
    const float* headCtx[3] = {ctxL, ctxR, ctxM};
    for (int hd = 0; hd < 3; ++hd) {
        colmatvec<<<4, 256, 0, stream>>>(headW[hd], Y, U, H_SZ, C_SZ);          // u = Wᵀ h₁
        dotk<<<1, 256, 0, stream>>>(headB[hd], Y, H_SZ, BQ);                    // bq = b·h₁
        wmma_gemv<<<GS, GEMV_BLOCK, 0, stream>>>(headCtx[hd], U, nullptr, nullptr, BQ,
                                                 Ebuf, S_SZ, C_SZ, 0, 0);       // e = ctx·u + bq
        softmax2048<<<1, 256, 0, stream>>>(Ebuf);                               // a = softmax(e)
        colmatvec<<<4, 256, 0, stream>>>(headCtx[hd], Ebuf, Y + (hd + 1) * C_SZ,
                                         S_SZ, C_SZ);                           // aᵀ·ctx
    }

    // ---- concat projection: tanh(W_concat·y + b_concat)
    wmma_gemv<<<GH, GEMV_BLOCK, 0, stream>>>(W_concat, Y, b_concat, nullptr, nullptr,
                                             COUT, H_SZ, 4 * H_SZ, 0, 1);

    // ---- output projection: logits = W_out·concat_out + b_out  (dominant, 206 MB stream)
    wmma_gemv<<<GV, GEMV_BLOCK, 0, stream>>>(W_out, COUT, b_out, nullptr, nullptr,
                                             out, V_SZ, H_SZ, 0, 0);
}